// MultiRelGraphTransformer_17205638988386
// MI455X (gfx1250) — compile-verified
//
#include <hip/hip_runtime.h>

#define B_ 4
#define N_ 20000
#define Dn_ 64
#define Dm_ 128
#define E_ 160000

typedef float v2f __attribute__((ext_vector_type(2)));
typedef float v8f __attribute__((ext_vector_type(8)));

__device__ __forceinline__ v8f wmma4(v2f a, v2f b, v8f c) {
  // D = A(16x4 f32) * B(4x16 f32) + C(16x16 f32)
  return __builtin_amdgcn_wmma_f32_16x16x4_f32(false, a, false, b, (short)0, c,
                                               false, false);
}

// ---------------------------------------------------------------------------
// Kernel 1: input projection  H[80000,128] = X[80000,64] @ W[64,128] + b
// block = 128 threads (4 waves), 16 rows per block, each wave does 2 col tiles
// ---------------------------------------------------------------------------
__global__ __launch_bounds__(128)
void proj_kernel(const float* __restrict__ X, const float* __restrict__ W,
                 const float* __restrict__ bias, float* __restrict__ H) {
  const int lane = threadIdx.x & 31;
  const int wave = threadIdx.x >> 5;
  const int m    = lane & 15;
  const int kh   = lane >> 4;            // 0: K pair {0,1}, 1: K pair {2,3}
  const int rowbase = blockIdx.x * 16;

  // A fragments for all of K=64 (16 steps of K=4)
  v2f a[16];
  const float* xr = X + (size_t)(rowbase + m) * Dn_;
#pragma unroll
  for (int kk = 0; kk < 16; ++kk) {
    int k0 = kk * 4 + kh * 2;
    a[kk].x = xr[k0];
    a[kk].y = xr[k0 + 1];
  }

#pragma unroll
  for (int t = 0; t < 2; ++t) {
    int col = (wave * 2 + t) * 16;
    v8f c = {0.f, 0.f, 0.f, 0.f, 0.f, 0.f, 0.f, 0.f};
#pragma unroll
    for (int kk = 0; kk < 16; ++kk) {
      int k0 = kk * 4 + kh * 2;
      v2f bf;
      bf.x = W[k0 * Dm_ + col + m];
      bf.y = W[(k0 + 1) * Dm_ + col + m];
      c = wmma4(a[kk], bf, c);
    }
    int rowadd = kh * 8;  // lanes 16-31 hold rows v+8
#pragma unroll
    for (int v = 0; v < 8; ++v) {
      H[(size_t)(rowbase + v + rowadd) * Dm_ + col + m] = c[v] + bias[col + m];
    }
  }
}

// ---------------------------------------------------------------------------
// Kernel 2: zero the aggregation buffer
// ---------------------------------------------------------------------------
__global__ __launch_bounds__(256)
void zero_kernel(float4* __restrict__ p, int n4) {
  int i = blockIdx.x * 256 + threadIdx.x;
  if (i < n4) p[i] = make_float4(0.f, 0.f, 0.f, 0.f);
}

// ---------------------------------------------------------------------------
// Kernel 3: per-edge messages + scatter-add for one layer, all 3 relations
// grid = (E/16, 3); block = 128 threads; one block: 16 edges x 4 batches
// ---------------------------------------------------------------------------
__global__ __launch_bounds__(128)
void edge_kernel(const float* __restrict__ H, float* __restrict__ agg,
                 const float* __restrict__ node_W, const float* __restrict__ node_b,
                 const float* __restrict__ edge_W, const float* __restrict__ edge_b,
                 const float* __restrict__ ea0, const float* __restrict__ ea1,
                 const int* __restrict__ ei0, const int* __restrict__ ei1,
                 const int* __restrict__ ei2, int layer) {
  __shared__ float T[16][132];    // bias + edge-attr term, padded stride
  __shared__ float Hs[16][132];   // gathered source rows, padded stride
  __shared__ int s_src[16];
  __shared__ int s_dst[16];

  const int r     = blockIdx.y;
  const int ebase = blockIdx.x * 16;
  const int tid   = threadIdx.x;
  const int lane  = tid & 31;
  const int wave  = tid >> 5;
  const int m     = lane & 15;
  const int kh    = lane >> 4;

  const int* ei = (r == 0) ? ei0 : (r == 1) ? ei1 : ei2;
  if (tid < 16) {
    s_src[tid] = ei[ebase + tid];        // row 0 of [2,E]
    s_dst[tid] = ei[E_ + ebase + tid];   // row 1 of [2,E]
  }

  const float* nb = node_b + (size_t)(layer * 3 + r) * Dm_;

  if (r < 2) {
    // T = ea[16x16] @ edge_W[16x128] + edge_b + node_b   (batch-shared)
    const float* ea = (r == 0) ? ea0 : ea1;
    const float* Wq = edge_W + (size_t)(layer * 2 + r) * 16 * Dm_;
    const float* eb = edge_b + (size_t)(layer * 2 + r) * Dm_;
    v2f a[4];
    const float* er = ea + (size_t)(ebase + m) * 16;
#pragma unroll
    for (int kk = 0; kk < 4; ++kk) {
      int k0 = kk * 4 + kh * 2;
      a[kk].x = er[k0];
      a[kk].y = er[k0 + 1];
    }
#pragma unroll
    for (int t = 0; t < 2; ++t) {
      int col = (wave * 2 + t) * 16;
      v8f c = {0.f, 0.f, 0.f, 0.f, 0.f, 0.f, 0.f, 0.f};
#pragma unroll
      for (int kk = 0; kk < 4; ++kk) {
        int k0 = kk * 4 + kh * 2;
        v2f bf;
        bf.x = Wq[k0 * Dm_ + col + m];
        bf.y = Wq[(k0 + 1) * Dm_ + col + m];
        c = wmma4(a[kk], bf, c);
      }
      int rowadd = kh * 8;
#pragma unroll
      for (int v = 0; v < 8; ++v) {
        T[v + rowadd][col + m] = c[v] + nb[col + m] + eb[col + m];
      }
    }
  } else {
    // relation 2 has no edge attrs: T = node_b broadcast
    for (int i = tid; i < 16 * Dm_; i += 128) {
      int row = i >> 7;
      int cc  = i & 127;
      T[row][cc] = nb[cc];
    }
  }
  __syncthreads();

  const float* Wn = node_W + (size_t)(layer * 3 + r) * Dm_ * Dm_;

  for (int b = 0; b < B_; ++b) {
    // cooperative gather of 16 source rows into LDS (float4, coalesced)
#pragma unroll
    for (int j = 0; j < 4; ++j) {
      int idx = tid + 128 * j;     // 0..511
      int row = idx >> 5;
      int c4  = idx & 31;
      const float* src = H + ((size_t)b * N_ + s_src[row]) * Dm_ + c4 * 4;
      float4 v = *(const float4*)src;
      *(float4*)&Hs[row][c4 * 4] = v;
    }
    __syncthreads();

#pragma unroll
    for (int t = 0; t < 2; ++t) {
      int col = (wave * 2 + t) * 16;
      v8f c = {0.f, 0.f, 0.f, 0.f, 0.f, 0.f, 0.f, 0.f};
#pragma unroll 8
      for (int kk = 0; kk < 32; ++kk) {     // K = 128
        int k0 = kk * 4 + kh * 2;
        v2f af;
        af.x = Hs[m][k0];
        af.y = Hs[m][k0 + 1];
        v2f bf;
        bf.x = Wn[k0 * Dm_ + col + m];
        bf.y = Wn[(k0 + 1) * Dm_ + col + m];
        c = wmma4(af, bf, c);
      }
      int rowadd = kh * 8;
#pragma unroll
      for (int v = 0; v < 8; ++v) {
        int mr = v + rowadd;
        float val = c[v] + T[mr][col + m];
        atomicAdd(&agg[((size_t)b * N_ + s_dst[mr]) * Dm_ + col + m], val);
      }
    }
    __syncthreads();  // Hs reused next batch
  }
}

// ---------------------------------------------------------------------------
// Kernel 4: H = LayerNorm(H + relu(agg)) * g + b   (in-place on H)
// one wave32 per 128-wide row, 4 rows per block
// ---------------------------------------------------------------------------
__global__ __launch_bounds__(128)
void ln_kernel(float* __restrict__ H, const float* __restrict__ agg,
               const float* __restrict__ ln_g, const float* __restrict__ ln_b,
               int layer) {
  const int lane = threadIdx.x & 31;
  const int wave = threadIdx.x >> 5;
  const size_t row = (size_t)blockIdx.x * 4 + wave;

  float* hp = H + row * Dm_;
  const float* ap = agg + row * Dm_;
  float4 x = *(const float4*)(hp + lane * 4);
  float4 a = *(const float4*)(ap + lane * 4);
  float4 y;
  y.x = x.x + fmaxf(a.x, 0.f);
  y.y = x.y + fmaxf(a.y, 0.f);
  y.z = x.z + fmaxf(a.z, 0.f);
  y.w = x.w + fmaxf(a.w, 0.f);

  float s = y.x + y.y + y.z + y.w;
  float q = y.x * y.x + y.y * y.y + y.z * y.z + y.w * y.w;
#pragma unroll
  for (int off = 16; off >= 1; off >>= 1) {   // wave32 reduction
    s += __shfl_xor(s, off, 32);
    q += __shfl_xor(q, off, 32);
  }
  float mean = s * (1.0f / Dm_);
  float var  = q * (1.0f / Dm_) - mean * mean;
  float rs   = rsqrtf(var + 1e-5f);

  const float* g  = ln_g + (size_t)layer * Dm_ + lane * 4;
  const float* bb = ln_b + (size_t)layer * Dm_ + lane * 4;
  float4 o;
  o.x = (y.x - mean) * rs * g[0] + bb[0];
  o.y = (y.y - mean) * rs * g[1] + bb[1];
  o.z = (y.z - mean) * rs * g[2] + bb[2];
  o.w = (y.w - mean) * rs * g[3] + bb[3];
  *(float4*)(hp + lane * 4) = o;
}

// ---------------------------------------------------------------------------
extern "C" void kernel_launch(void* const* d_in, const int* in_sizes, int n_in,
                              void* d_out, int out_size, void* d_ws, size_t ws_size,
                              hipStream_t stream) {
  const float* node_feat = (const float*)d_in[0];
  const float* in_W      = (const float*)d_in[1];
  const float* in_b      = (const float*)d_in[2];
  const float* node_W    = (const float*)d_in[3];
  const float* node_b    = (const float*)d_in[4];
  const float* edge_W    = (const float*)d_in[5];
  const float* edge_b    = (const float*)d_in[6];
  const float* ln_g      = (const float*)d_in[7];
  const float* ln_b      = (const float*)d_in[8];
  const float* ea0       = (const float*)d_in[9];
  const float* ea1       = (const float*)d_in[10];
  const int*   ei0       = (const int*)d_in[11];
  const int*   ei1       = (const int*)d_in[12];
  const int*   ei2       = (const int*)d_in[13];

  float* H   = (float*)d_out;   // H lives in d_out (41 MB), LN is in-place
  float* agg = (float*)d_ws;    // 41 MB scratch

  proj_kernel<<<dim3((B_ * N_) / 16), dim3(128), 0, stream>>>(node_feat, in_W,
                                                              in_b, H);
  const int n4 = (B_ * N_ * Dm_) / 4;
  for (int l = 0; l < 2; ++l) {
    zero_kernel<<<dim3((n4 + 255) / 256), dim3(256), 0, stream>>>((float4*)agg,
                                                                  n4);
    edge_kernel<<<dim3(E_ / 16, 3), dim3(128), 0, stream>>>(
        H, agg, node_W, node_b, edge_W, edge_b, ea0, ea1, ei0, ei1, ei2, l);
    ln_kernel<<<dim3((B_ * N_) / 4), dim3(128), 0, stream>>>(H, agg, ln_g,
                                                             ln_b, l);
  }
}